// EncodingLoss_2473901162906
// MI455X (gfx1250) — compile-verified
//
#include <hip/hip_runtime.h>

// ---------------------------------------------------------------------------
// EncodingLoss for MI455X (gfx1250, wave32).
//   obs:      [4096, 64]  f32
//   encoding: [4096, 128] f32
//   loss = mean over all 4096^2 pairs of relu(sign * enc_dist + 1)
//   where sign = -1 if obs_dist > 0.01 else +1, dists use the gram expansion
//   with torch pairwise_distance eps folded in.
// Strategy: fused WMMA gram tiles (V_WMMA_F32_16X16X4_F32), never materialize
// the NxN matrices.
// ---------------------------------------------------------------------------

#define NROWS 4096
#define DOBS 64
#define DENC 128
#define EPSQ 1e-6f
#define MARGIN_F 1.0f
#define THRESH_SQ 1e-4f /* 0.01^2 */

typedef __attribute__((ext_vector_type(2))) float v2f;
typedef __attribute__((ext_vector_type(8))) float v8f;

__device__ __forceinline__ float wave_reduce_sum(float v) {
  // wave32 butterfly reduction
#pragma unroll
  for (int off = 16; off > 0; off >>= 1) v += __shfl_down(v, off, 32);
  return v;
}

// ws float layout: [0] = global accumulator
//                  [64 .. 64+N)        sq_norm(obs)
//                  [64+N .. 64+2N)     row_sum(obs)
//                  [64+2N .. 64+3N)    sq_norm(enc)
//                  [64+3N .. 64+4N)    row_sum(enc)

__global__ __launch_bounds__(256) void stats_kernel(
    const float* __restrict__ obs, const float* __restrict__ enc,
    float* __restrict__ ws) {
  if (blockIdx.x == 0 && threadIdx.x == 0) ws[0] = 0.0f;  // reset accumulator

  const int wave = threadIdx.x >> 5;
  const int lane = threadIdx.x & 31;
  const int row = blockIdx.x * 8 + wave;
  if (row >= NROWS) return;

  // obs row: 64 floats -> 2 per lane
  const float* o = obs + (size_t)row * DOBS;
  float a0 = o[lane], a1 = o[lane + 32];
  float s = a0 + a1;
  float ss = a0 * a0 + a1 * a1;
  s = wave_reduce_sum(s);
  ss = wave_reduce_sum(ss);

  // enc row: 128 floats -> 4 per lane
  const float* e = enc + (size_t)row * DENC;
  float es = 0.f, ess = 0.f;
#pragma unroll
  for (int t = 0; t < 4; ++t) {
    float v = e[lane + 32 * t];
    es += v;
    ess += v * v;
  }
  es = wave_reduce_sum(es);
  ess = wave_reduce_sum(ess);

  if (lane == 0) {
    float* sqnO = ws + 64;
    float* rsO = ws + 64 + NROWS;
    float* sqnE = ws + 64 + 2 * NROWS;
    float* rsE = ws + 64 + 3 * NROWS;
    sqnO[row] = ss;
    rsO[row] = s;
    sqnE[row] = ess;
    rsE[row] = es;
  }
}

__global__ __launch_bounds__(256) void tile_kernel(
    const float* __restrict__ obs, const float* __restrict__ enc,
    const float* __restrict__ stats, float* __restrict__ acc_out) {
  __shared__ float blk_acc;
  if (threadIdx.x == 0) blk_acc = 0.0f;
  __syncthreads();

  const int wave = threadIdx.x >> 5;
  const int lane = threadIdx.x & 31;
  const int tile = blockIdx.x * 8 + wave;  // 0 .. 65535
  const int tI = tile >> 8;                // row tile 0..255
  const int tJ = tile & 255;               // col tile 0..255

  const int m = lane & 15;   // M (A) / N (B) index held by this lane
  const int kg = lane >> 4;  // K group: lane holds K = 2*kg + {0,1}

  // ---- obs gram tile: 16 x WMMA f32 16x16x4 ----
  const float* Ao = obs + (size_t)(tI * 16 + m) * DOBS + 2 * kg;
  const float* Bo = obs + (size_t)(tJ * 16 + m) * DOBS + 2 * kg;
  v8f accO = {};
#pragma unroll
  for (int k = 0; k < DOBS; k += 4) {
    v2f a = *(const v2f*)(Ao + k);
    v2f b = *(const v2f*)(Bo + k);
    accO = __builtin_amdgcn_wmma_f32_16x16x4_f32(false, a, false, b, (short)0,
                                                 accO, false, false);
  }

  // ---- encoding gram tile: 32 x WMMA f32 16x16x4 ----
  const float* Ae = enc + (size_t)(tI * 16 + m) * DENC + 2 * kg;
  const float* Be = enc + (size_t)(tJ * 16 + m) * DENC + 2 * kg;
  v8f accE = {};
#pragma unroll
  for (int k = 0; k < DENC; k += 4) {
    v2f a = *(const v2f*)(Ae + k);
    v2f b = *(const v2f*)(Be + k);
    accE = __builtin_amdgcn_wmma_f32_16x16x4_f32(false, a, false, b, (short)0,
                                                 accE, false, false);
  }

  // ---- fused distance + mask + hinge on the C/D layout ----
  // C/D 16x16 f32: VGPR r, lane L -> row M = r + 8*(L/16), col N = L%16
  const float* sqnO = stats;
  const float* rsO = stats + NROWS;
  const float* sqnE = stats + 2 * NROWS;
  const float* rsE = stats + 3 * NROWS;

  const int j = tJ * 16 + m;
  const float sqnOj = sqnO[j], rsOj = rsO[j];
  const float sqnEj = sqnE[j], rsEj = rsE[j];
  const float floorO = (float)DOBS * EPSQ * EPSQ;
  const float floorE = (float)DENC * EPSQ * EPSQ;

  float partial = 0.0f;
#pragma unroll
  for (int r = 0; r < 8; ++r) {
    const int i = tI * 16 + kg * 8 + r;
    float sqO = sqnO[i] + sqnOj - 2.0f * accO[r] +
                2.0f * EPSQ * (rsO[i] - rsOj) + floorO;
    sqO = fmaxf(sqO, floorO);
    const bool negative = sqO > THRESH_SQ;  // obs_dist > 0.01

    float sqE = sqnE[i] + sqnEj - 2.0f * accE[r] +
                2.0f * EPSQ * (rsE[i] - rsEj) + floorE;
    const float dE = sqrtf(fmaxf(sqE, floorE));
    const float signedv = negative ? -dE : dE;
    partial += fmaxf(signedv + MARGIN_F, 0.0f);
  }

  partial = wave_reduce_sum(partial);
  if (lane == 0) atomicAdd(&blk_acc, partial);
  __syncthreads();
  if (threadIdx.x == 0) atomicAdd(acc_out, blk_acc);
}

__global__ void finalize_kernel(const float* __restrict__ acc,
                                float* __restrict__ out) {
  out[0] = acc[0] * (1.0f / ((float)NROWS * (float)NROWS));
}

extern "C" void kernel_launch(void* const* d_in, const int* in_sizes, int n_in,
                              void* d_out, int out_size, void* d_ws,
                              size_t ws_size, hipStream_t stream) {
  (void)in_sizes;
  (void)n_in;
  (void)out_size;
  (void)ws_size;
  const float* obs = (const float*)d_in[0];
  const float* enc = (const float*)d_in[1];
  float* w = (float*)d_ws;

  // 1) per-row stats (+ accumulator reset): 4096 rows, 8 waves per block
  stats_kernel<<<NROWS / 8, 256, 0, stream>>>(obs, enc, w);

  // 2) fused WMMA gram tiles + hinge reduction: 256x256 tiles, 8 waves/block
  const int n_tiles = (NROWS / 16) * (NROWS / 16);  // 65536
  tile_kernel<<<n_tiles / 8, 256, 0, stream>>>(obs, enc, w + 64, w);

  // 3) mean
  finalize_kernel<<<1, 1, 0, stream>>>(w, (float*)d_out);
}